// GraphAutoencoder_10823317586422
// MI455X (gfx1250) — compile-verified
//
#include <hip/hip_runtime.h>
#include <hip/hip_bf16.h>

typedef __attribute__((ext_vector_type(2))) float v2f;
typedef __attribute__((ext_vector_type(8))) float v8f;

// ---------------------------------------------------------------- utilities
__global__ void fill_kernel(float* __restrict__ p, long n, float v) {
    long t = (long)blockIdx.x * blockDim.x + threadIdx.x;
    if (t < n) p[t] = v;
}

__global__ void degree_kernel(const int* __restrict__ dst, float* __restrict__ deg, int nE) {
    int e = blockIdx.x * blockDim.x + threadIdx.x;
    if (e < nE) atomicAdd(&deg[dst[e]], 1.0f);
}

__global__ void invsqrt_kernel(float* __restrict__ dis, int n) {
    int i = blockIdx.x * blockDim.x + threadIdx.x;
    if (i < n) dis[i] = 1.0f / sqrtf(dis[i] + 1.0f);   // deg includes self loop
}

// ------------------------------------------------- WMMA fp32 GEMM: H = X @ W
// One wave32 computes a 16-row stripe across ALL output columns with
// V_WMMA_F32_16X16X4_F32, reusing the A fragment across TN column tiles.
// Requires nRows % 16 == 0 (N = 100000 = 6250*16).
// A 16x4 frag: lanes 0-15 -> K = k0,k0+1 ; lanes 16-31 -> K = k0+2,k0+3 (M = lane&15)
// B 4x16 frag: same K split across lane halves (N = lane&15)
// D 16x16:     VGPR r -> M = r + 8*(lane>>4), N = lane&15
template <int FIN, int FOUT>
__global__ void __launch_bounds__(128)
gemm_wmma_kernel(const float* __restrict__ X, const float* __restrict__ W,
                 float* __restrict__ H, int tilesM) {
    constexpr int TN = (FOUT + 15) / 16;
    const int tm = blockIdx.x * blockDim.y + threadIdx.y;
    if (tm >= tilesM) return;

    const int lane = threadIdx.x;      // 0..31
    const int half = lane >> 4;        // 0 or 1
    const int lq   = lane & 15;
    const int row  = tm * 16 + lq;     // A row this lane supplies

    v8f acc[TN];
#pragma unroll
    for (int t = 0; t < TN; ++t) acc[t] = v8f{};

#pragma unroll
    for (int k0 = 0; k0 < FIN; k0 += 4) {
        const int ka = k0 + 2 * half;  // even -> 8B-aligned pair
        v2f a;
        if constexpr (FIN % 4 == 0) {
            a = *reinterpret_cast<const v2f*>(X + (size_t)row * FIN + ka);
        } else {
            // FIN == 3: zero-pad K=3 branchlessly (clamped dummy load + select)
            const size_t base = (size_t)row * FIN;
            const float ax = X[base + ka];                    // K0 / K2, always valid
            const float ay = X[base + ka + (half ? 0 : 1)];   // K1, or dummy re-read of K2
            a.x = ax;
            a.y = half ? 0.0f : ay;
        }
#pragma unroll
        for (int tn = 0; tn < TN; ++tn) {
            const int col = tn * 16 + lq;
            v2f b;
            if constexpr (FOUT % 16 == 0) {
                b.x = W[(size_t)ka * FOUT + col];
                b.y = W[(size_t)(ka + 1) * FOUT + col];
            } else {
                // FOUT == 3: clamp address, select zero for padded columns
                const int ccol = col < FOUT ? col : FOUT - 1;
                const float bx = W[(size_t)ka * FOUT + ccol];
                const float by = W[(size_t)(ka + 1) * FOUT + ccol];
                b.x = col < FOUT ? bx : 0.0f;
                b.y = col < FOUT ? by : 0.0f;
            }
            acc[tn] = __builtin_amdgcn_wmma_f32_16x16x4_f32(
                /*neg_a=*/false, a, /*neg_b=*/false, b,
                /*c_mod=*/(short)0, acc[tn], /*reuse_a=*/false, /*reuse_b=*/false);
        }
    }

#pragma unroll
    for (int tn = 0; tn < TN; ++tn) {
        const int col = tn * 16 + lq;
        if (FOUT % 16 == 0 || col < FOUT) {
#pragma unroll
            for (int r = 0; r < 8; ++r) {
                const int m = tm * 16 + r + 8 * half;
                H[(size_t)m * FOUT + col] = acc[tn][r];
            }
        }
    }
}

// ------------------------------------- edge scatter: agg[dst] += H[src]*norm
// F multiple of 4: thread handles one float4 chunk; for F=128 one wave == one edge.
template <int F>
__global__ void scatter_edges_kernel(const float* __restrict__ H,
                                     const int* __restrict__ src,
                                     const int* __restrict__ dst,
                                     const float* __restrict__ dis,
                                     float* __restrict__ agg, int nE) {
    constexpr int CH = F / 4;
    long tid = (long)blockIdx.x * blockDim.x + threadIdx.x;
    if (tid >= (long)nE * CH) return;
    const int e  = (int)(tid / CH);
    const int ch = (int)(tid % CH);
    const int s = src[e], d = dst[e];
    const float norm = dis[s] * dis[d];
    const float4 hv = *reinterpret_cast<const float4*>(H + (size_t)s * F + ch * 4);
    float* ap = agg + (size_t)d * F + ch * 4;
    atomicAdd(ap + 0, hv.x * norm);
    atomicAdd(ap + 1, hv.y * norm);
    atomicAdd(ap + 2, hv.z * norm);
    atomicAdd(ap + 3, hv.w * norm);
}

__global__ void scatter_edges3_kernel(const float* __restrict__ H,
                                      const int* __restrict__ src,
                                      const int* __restrict__ dst,
                                      const float* __restrict__ dis,
                                      float* __restrict__ agg, int nE) {
    int e = blockIdx.x * blockDim.x + threadIdx.x;
    if (e >= nE) return;
    const int s = src[e], d = dst[e];
    const float norm = dis[s] * dis[d];
    const float* hp = H + (size_t)s * 3;
    float* ap = agg + (size_t)d * 3;
    atomicAdd(ap + 0, hp[0] * norm);
    atomicAdd(ap + 1, hp[1] * norm);
    atomicAdd(ap + 2, hp[2] * norm);
}

// ------------------ finalize: out = agg + H*dis^2 + bias (optionally ReLU)
__global__ void finalize_kernel(const float* __restrict__ agg, const float* __restrict__ H,
                                const float* __restrict__ dis, const float* __restrict__ bias,
                                float* __restrict__ out, int n, int F, int doRelu) {
    long t = (long)blockIdx.x * blockDim.x + threadIdx.x;
    if (t >= (long)n * F) return;
    const int i = (int)(t / F);
    const int f = (int)(t % F);
    const float di = dis[i];
    float v = agg[t] + H[t] * di * di + bias[f];
    if (doRelu) v = fmaxf(v, 0.0f);
    out[t] = v;
}

// ---------------------------------------------------------------- driver
static inline long cdivl(long a, long b) { return (a + b - 1) / b; }

extern "C" void kernel_launch(void* const* d_in, const int* in_sizes, int n_in,
                              void* d_out, int out_size, void* d_ws, size_t ws_size,
                              hipStream_t stream) {
    const float* x  = (const float*)d_in[0];
    const int*   ei = (const int*)d_in[1];
    const float* W1 = (const float*)d_in[2];
    const float* b1 = (const float*)d_in[3];
    const float* W2 = (const float*)d_in[4];
    const float* b2 = (const float*)d_in[5];
    const float* W3 = (const float*)d_in[6];
    const float* b3 = (const float*)d_in[7];
    const float* W4 = (const float*)d_in[8];
    const float* b4 = (const float*)d_in[9];

    const int N = in_sizes[0] / 3;
    const int E = in_sizes[1] / 2;
    const int* srcv = ei;
    const int* dstv = ei + E;

    float* recon  = (float*)d_out;                 // [N, 3]
    float* latent = recon + (size_t)N * 3;         // [N, 64]

    // workspace layout (floats): dis[N] | H[N*128] | AGG[N*128] | ACT[N*128]
    float* ws  = (float*)d_ws;
    float* dis = ws;
    float* H   = dis + N;
    float* AGG = H   + (size_t)N * 128;
    float* ACT = AGG + (size_t)N * 128;

    const int TB = 256;
    dim3 gblk(32, 4);                 // 4 waves per block; one 16-row stripe each
    const int tilesM = N / 16;        // N = 100000 -> 6250 exact
    const int gemmBlocks = (int)cdivl(tilesM, 4);

    // ---- degrees -> dis = 1/sqrt(deg+1), in place
    fill_kernel<<<(int)cdivl(N, TB), TB, 0, stream>>>(dis, N, 0.0f);
    degree_kernel<<<(int)cdivl(E, TB), TB, 0, stream>>>(dstv, dis, E);
    invsqrt_kernel<<<(int)cdivl(N, TB), TB, 0, stream>>>(dis, N);

    // ---- Layer 1: [N,3] @ [3,128], ReLU -> ACT
    {
        gemm_wmma_kernel<3, 128><<<gemmBlocks, gblk, 0, stream>>>(x, W1, H, tilesM);
        fill_kernel<<<(int)cdivl((long)N * 128, TB), TB, 0, stream>>>(AGG, (long)N * 128, 0.0f);
        scatter_edges_kernel<128><<<(int)cdivl((long)E * 32, TB), TB, 0, stream>>>(H, srcv, dstv, dis, AGG, E);
        finalize_kernel<<<(int)cdivl((long)N * 128, TB), TB, 0, stream>>>(AGG, H, dis, b1, ACT, N, 128, 1);
    }

    // ---- Layer 2: [N,128] @ [128,64] -> latent (d_out), no ReLU
    {
        gemm_wmma_kernel<128, 64><<<gemmBlocks, gblk, 0, stream>>>(ACT, W2, H, tilesM);
        fill_kernel<<<(int)cdivl((long)N * 64, TB), TB, 0, stream>>>(AGG, (long)N * 64, 0.0f);
        scatter_edges_kernel<64><<<(int)cdivl((long)E * 16, TB), TB, 0, stream>>>(H, srcv, dstv, dis, AGG, E);
        finalize_kernel<<<(int)cdivl((long)N * 64, TB), TB, 0, stream>>>(AGG, H, dis, b2, latent, N, 64, 0);
    }

    // ---- Layer 3: [N,64] @ [64,128], ReLU -> ACT
    {
        gemm_wmma_kernel<64, 128><<<gemmBlocks, gblk, 0, stream>>>(latent, W3, H, tilesM);
        fill_kernel<<<(int)cdivl((long)N * 128, TB), TB, 0, stream>>>(AGG, (long)N * 128, 0.0f);
        scatter_edges_kernel<128><<<(int)cdivl((long)E * 32, TB), TB, 0, stream>>>(H, srcv, dstv, dis, AGG, E);
        finalize_kernel<<<(int)cdivl((long)N * 128, TB), TB, 0, stream>>>(AGG, H, dis, b3, ACT, N, 128, 1);
    }

    // ---- Layer 4: [N,128] @ [128,3] -> recon (d_out), no ReLU
    {
        gemm_wmma_kernel<128, 3><<<gemmBlocks, gblk, 0, stream>>>(ACT, W4, H, tilesM);
        fill_kernel<<<(int)cdivl((long)N * 3, TB), TB, 0, stream>>>(AGG, (long)N * 3, 0.0f);
        scatter_edges3_kernel<<<(int)cdivl(E, TB), TB, 0, stream>>>(H, srcv, dstv, dis, AGG, E);
        finalize_kernel<<<(int)cdivl((long)N * 3, TB), TB, 0, stream>>>(AGG, H, dis, b4, recon, N, 3, 0);
    }
}